// TreeKernel_29068338659390
// MI455X (gfx1250) — compile-verified
//
#include <hip/hip_runtime.h>

typedef float v2f __attribute__((ext_vector_type(2)));
typedef float v8f __attribute__((ext_vector_type(8)));

#define TK_EPS       1e-6f
#define TK_MAX_NODES 15
#define TK_NUM_ATOMS 3
#define TK_PI        3.14159265358979323846f

__device__ __forceinline__ void ld8(float dst[8], const float src[8]) {
#pragma unroll
  for (int r = 0; r < 8; ++r) dst[r] = src[r];
}

__global__ __launch_bounds__(256) void tree_eval_wmma(
    const float* __restrict__ x, const float* __restrict__ y,
    const int* __restrict__ post_expr, const int* __restrict__ post_level,
    const unsigned char* __restrict__ is_op, int n_lib,
    const float* __restrict__ params,
    float* __restrict__ out, int D1, int D2) {
  const int lane = threadIdx.x & 31;
  const int wave = threadIdx.x >> 5;
  const int tilesJ = D2 >> 4;
  const int tilesI = D1 >> 4;
  const int tile = blockIdx.x * (blockDim.x >> 5) + wave;
  if (tile >= tilesI * tilesJ) return;   // wave-uniform: EXEC stays all-ones
  const int ti = tile / tilesJ;
  const int tj = tile - ti * tilesJ;
  const int i0 = ti << 4;
  const int j0 = tj << 4;
  const int lo = lane & 15;
  const int hi = lane >> 4;

  // ---- Build the two bilinear interaction tiles with V_WMMA_F32_16X16X4_F32.
  // A (16x4 f32): lanes 0-15 hold K=0 (v.x) and K=1 (v.y); lanes 16-31 hold K=2,3.
  // B (4x16 f32): VGPR0 = rows K=0 (lanes 0-15) / K=2 (lanes 16-31); VGPR1 = K=1 / K=3.
  const float xl = x[i0 + lo];
  const float yl = y[j0 + lo];

  // dx = x_i - y_j  ==  [x, -1, 0, 0] x [1; y; 0; 0]
  v2f a_dx, b_dx, a_xy, b_xy;
  a_dx.x = hi ? 0.0f : xl;   a_dx.y = hi ? 0.0f : -1.0f;
  b_dx.x = hi ? 0.0f : 1.0f; b_dx.y = hi ? 0.0f : yl;
  // xy = x_i * y_j  ==  [x, 0, 0, 0] x [y; 0; 0; 0]
  a_xy.x = hi ? 0.0f : xl;   a_xy.y = 0.0f;
  b_xy.x = hi ? 0.0f : yl;   b_xy.y = 0.0f;

  v8f zeroC = {};
  v8f dxm = __builtin_amdgcn_wmma_f32_16x16x4_f32(
      false, a_dx, false, b_dx, (short)0, zeroC, false, false);
  v8f xym = __builtin_amdgcn_wmma_f32_16x16x4_f32(
      false, a_xy, false, b_xy, (short)0, zeroC, false, false);

  // ---- Generic post-order stack interpreter (STACK_SIZE = 4), stack in registers.
  float s0[8], s1[8], s2[8], s3[8];
#pragma unroll
  for (int r = 0; r < 8; ++r) { s0[r] = 0.f; s1[r] = 0.f; s2[r] = 0.f; s3[r] = 0.f; }
  int ptr = 0;

#define STK_WRITE(idx, val) do { switch ((idx) & 3) { \
    case 0: ld8(s0, val); break; case 1: ld8(s1, val); break; \
    case 2: ld8(s2, val); break; default: ld8(s3, val); break; } } while (0)
#define STK_READ(idx, dst) do { switch ((idx) & 3) { \
    case 0: ld8(dst, s0); break; case 1: ld8(dst, s1); break; \
    case 2: ld8(dst, s2); break; default: ld8(dst, s3); break; } } while (0)

  for (int n = 0; n < TK_MAX_NODES; ++n) {
    const int nv = post_expr[n];            // wave-uniform scalar loads
    if (nv < 0) continue;
    int ci = nv; if (ci < 0) ci = 0; if (ci > n_lib - 1) ci = n_lib - 1;
    const bool isop = is_op[ci] != 0;
    float k[8];
    if (!isop) {
      const int lvl = post_level[n];
      const float p0 = params[lvl * 3 + 0];
      const float p1 = params[lvl * 3 + 1];
      const float p2 = params[lvl * 3 + 2];
      int atom = nv; if (atom > TK_NUM_ATOMS - 1) atom = TK_NUM_ATOMS - 1;
      if (atom == 0) {                       // RBF: p0^2 * exp(-0.5*dx^2/(p1^2+eps))
        const float sc = -0.5f / (p1 * p1 + TK_EPS);
        const float a2 = p0 * p0;
#pragma unroll
        for (int r = 0; r < 8; ++r) {
          const float d = dxm[r];
          k[r] = a2 * __expf(d * d * sc);    // v_exp_f32 (TRANS, co-executes w/ VALU)
        }
      } else if (atom == 1) {                // linear: p0^2 * x*y + p1^2
        const float a2 = p0 * p0, b2 = p1 * p1;
#pragma unroll
        for (int r = 0; r < 8; ++r) k[r] = a2 * xym[r] + b2;
      } else {                               // periodic
        const float period = p2 * p2 + 1.0f;
        const float w = TK_PI / period;
        const float sc = -2.0f / (p1 * p1 + TK_EPS);
        const float a2 = p0 * p0;
#pragma unroll
        for (int r = 0; r < 8; ++r) {
          const float s = __sinf(dxm[r] * w);  // v_sin_f32; |arg| < ~20, accurate range
          k[r] = a2 * __expf(s * s * sc);
        }
      }
      STK_WRITE(ptr, k);
      ptr += 1;
    } else {
      float l[8], rr[8];
      STK_READ(ptr - 1, l);
      STK_READ(ptr - 2, rr);
      int op = nv - TK_NUM_ATOMS; if (op < 0) op = 0; if (op > 1) op = 1;
      if (op == 0) {
#pragma unroll
        for (int r = 0; r < 8; ++r) k[r] = l[r] + rr[r];
      } else {
#pragma unroll
        for (int r = 0; r < 8; ++r) k[r] = l[r] * rr[r];
      }
      STK_WRITE(ptr - 2, k);
      ptr -= 1;
    }
  }

  // ---- Store: C-layout VGPR r -> row i0 + r + 8*hi, lane -> col j0 + lo.
#pragma unroll
  for (int r = 0; r < 8; ++r)
    out[(i0 + r + 8 * hi) * D2 + j0 + lo] = s0[r];

#undef STK_WRITE
#undef STK_READ
}

extern "C" void kernel_launch(void* const* d_in, const int* in_sizes, int n_in,
                              void* d_out, int out_size, void* d_ws, size_t ws_size,
                              hipStream_t stream) {
  const float*         x          = (const float*)d_in[0];
  const float*         y          = (const float*)d_in[1];
  const int*           post_expr  = (const int*)d_in[2];
  const int*           post_level = (const int*)d_in[3];
  const unsigned char* is_op      = (const unsigned char*)d_in[4];  // jnp bool_ = 1 byte
  const float*         params     = (const float*)d_in[5];
  float*               out        = (float*)d_out;

  const int D1 = in_sizes[0];
  const int D2 = in_sizes[1];
  const int n_lib = in_sizes[4];

  const int tiles = (D1 >> 4) * (D2 >> 4);     // 16384 for 2048x2048
  const int wavesPerBlock = 8;                 // 256 threads = 8 wave32
  const int blocks = (tiles + wavesPerBlock - 1) / wavesPerBlock;

  tree_eval_wmma<<<blocks, 256, 0, stream>>>(x, y, post_expr, post_level,
                                             is_op, n_lib, params, out, D1, D2);
}